// SensitivitySpecificityLoss_57492432224511
// MI455X (gfx1250) — compile-verified
//
#include <hip/hip_runtime.h>
#include <hip/hip_bf16.h>

// ---------------------------------------------------------------------------
// SensitivitySpecificityLoss for MI455X (gfx1250, wave32)
//
// Pipeline:
//   1) zero_hist   : zero the 361-bin global confusion histogram in d_ws
//   2) argmax_hist : streaming argmax over 19 channels (float4/pixel-quad,
//                    non-temporal b128 loads), LDS histogram + global atomics
//   3) cm_loss     : 1-wave epilogue; row/col sums of the 19x19 confusion
//                    matrix computed with V_WMMA_F32_16X16X4_F32 (CM x ones
//                    and CM^T x ones), then sensitivity/specificity -> loss
//
// Memory-bound: ~670 MB streamed once => ~29 us at 23.3 TB/s.
// ---------------------------------------------------------------------------

typedef __attribute__((ext_vector_type(4))) float f32x4;
typedef __attribute__((ext_vector_type(4))) int   i32x4;
typedef __attribute__((ext_vector_type(2))) float v2f;
typedef __attribute__((ext_vector_type(8))) float v8f;

constexpr int kB    = 16;
constexpr int kC    = 19;
constexpr int kH    = 512;
constexpr int kW    = 1024;
constexpr int kHW   = kH * kW;        // 524288 = 2^19
constexpr int kHW4  = kHW / 4;        // 131072 = 2^17
constexpr int kNPix = kB * kHW;       // 8388608
constexpr int kN4   = kNPix / 4;      // 2097152
constexpr int kBins = kC * kC;        // 361

__global__ void zero_hist(unsigned int* __restrict__ g) {
  int i = threadIdx.x;
  if (i < kBins) g[i] = 0u;
}

__global__ __launch_bounds__(256)
void argmax_hist(const float* __restrict__ logits,
                 const int*   __restrict__ target,
                 unsigned int* __restrict__ ghist) {
  __shared__ unsigned int lh[kBins];
  for (int i = threadIdx.x; i < kBins; i += 256) lh[i] = 0u;
  __syncthreads();

  // One thread = 4 consecutive pixels (never crosses a batch boundary:
  // kHW divisible by 4). Element offset of (b,c,pix) = (b*kC + c)*kHW + pix.
  int gid = blockIdx.x * 256 + threadIdx.x;      // [0, kN4) exactly
  int b   = gid >> 17;                           // gid / kHW4
  int r   = gid & (kHW4 - 1);                    // gid % kHW4
  const f32x4* base = (const f32x4*)logits + (size_t)b * kC * kHW4 + r;

  // Channel 0 initializes the running max; strict '>' preserves the
  // first-occurrence-of-max semantics of jnp.argmax.
  f32x4 m = __builtin_nontemporal_load(base);
  int ax = 0, ay = 0, az = 0, aw = 0;
  #pragma unroll
  for (int c = 1; c < kC; ++c) {
    f32x4 v = __builtin_nontemporal_load(base + (size_t)c * kHW4);
    if (v.x > m.x) { m.x = v.x; ax = c; }
    if (v.y > m.y) { m.y = v.y; ay = c; }
    if (v.z > m.z) { m.z = v.z; az = c; }
    if (v.w > m.w) { m.w = v.w; aw = c; }
  }

  i32x4 t = __builtin_nontemporal_load((const i32x4*)target + gid);
  atomicAdd(&lh[t.x * kC + ax], 1u);
  atomicAdd(&lh[t.y * kC + ay], 1u);
  atomicAdd(&lh[t.z * kC + az], 1u);
  atomicAdd(&lh[t.w * kC + aw], 1u);

  __syncthreads();
  for (int i = threadIdx.x; i < kBins; i += 256) {
    unsigned int v = lh[i];
    if (v) atomicAdd(&ghist[i], v);
  }
}

// cm[i*19+j] where i = y_true (row), j = y_pred (col); zero-padded view.
#define CMV(i, j) ((((i) < kC) && ((j) < kC)) ? cm[(i) * kC + (j)] : 0.0f)

__global__ __launch_bounds__(32)
void cm_loss(const unsigned int* __restrict__ ghist, float* __restrict__ out) {
  __shared__ float cm[kBins];
  __shared__ float s_row[32];
  __shared__ float s_col[32];
  __shared__ float s_val[kC];

  int lane = threadIdx.x;
  for (int i = lane; i < kBins; i += 32) cm[i] = (float)ghist[i];
  __syncthreads();

  // A-matrix 16x4 f32 layout (ISA 7.12.2): lane L holds row M = L&15;
  // VGPR0 -> K = (L<16 ? 0 : 2), VGPR1 -> K = (L<16 ? 1 : 3).
  // B = all-ones 4x16, so D[m][n] = sum_k A[m][k] (rowsum replicated over N).
  // Feed CM for row sums and CM^T for col sums; two M-tiles cover 19 rows,
  // five K-chunks of 4 cover 19 columns (padding = 0). All guards are
  // selects, so EXEC stays all-ones around every WMMA.
  int mrow = lane & 15;
  int half = lane >> 4;
  v8f r0 = {}, r1 = {}, c0 = {}, c1 = {};
  const v2f ones = {1.0f, 1.0f};

  #pragma unroll
  for (int kc = 0; kc < 5; ++kc) {
    int k0 = kc * 4 + half * 2;
    int k1 = k0 + 1;
    v2f a;
    // rows 0..15 of CM
    a.x = CMV(mrow, k0); a.y = CMV(mrow, k1);
    r0 = __builtin_amdgcn_wmma_f32_16x16x4_f32(false, a, false, ones,
                                               (short)0, r0, false, false);
    // rows 16..18 of CM (rows 19..31 zero)
    a.x = CMV(mrow + 16, k0); a.y = CMV(mrow + 16, k1);
    r1 = __builtin_amdgcn_wmma_f32_16x16x4_f32(false, a, false, ones,
                                               (short)0, r1, false, false);
    // rows 0..15 of CM^T  -> column sums of CM
    a.x = CMV(k0, mrow); a.y = CMV(k1, mrow);
    c0 = __builtin_amdgcn_wmma_f32_16x16x4_f32(false, a, false, ones,
                                               (short)0, c0, false, false);
    // rows 16..18 of CM^T
    a.x = CMV(k0, mrow + 16); a.y = CMV(k1, mrow + 16);
    c1 = __builtin_amdgcn_wmma_f32_16x16x4_f32(false, a, false, ones,
                                               (short)0, c1, false, false);
  }

  // D layout (16x16 f32): VGPR r, lane L -> M = r + 8*(L>>4), N = L&15.
  // Result is constant across N, so lanes with N==0 (lanes 0 and 16)
  // publish all 8 accumulator registers.
  if (mrow == 0) {
    #pragma unroll
    for (int rr = 0; rr < 8; ++rr) {
      s_row[rr + 8 * half]      = r0[rr];
      s_row[16 + rr + 8 * half] = r1[rr];
      s_col[rr + 8 * half]      = c0[rr];
      s_col[16 + rr + 8 * half] = c1[rr];
    }
  }
  __syncthreads();

  if (lane < kC) {
    float tp = cm[lane * kC + lane];
    float fp = s_row[lane] - tp;                 // sum over axis=1 - tp
    float fn = s_col[lane] - tp;                 // sum over axis=0 - tp
    float tn = (float)kNPix - tp - fp - fn;      // total CM sum == #pixels
    float sens = (tp + 1e-6f) / (tp + fn + 1e-6f);
    float spec = (tn + 1e-6f) / (tn + fp + 1e-6f);
    s_val[lane] = 0.5f * sens + 0.5f * spec;     // WEIGHT = 0.5
  }
  __syncthreads();

  if (lane == 0) {
    float s = 0.0f;
    #pragma unroll
    for (int i = 0; i < kC; ++i) s += s_val[i];
    out[0] = 1.0f - s / 19.0f;
  }
}

extern "C" void kernel_launch(void* const* d_in, const int* in_sizes, int n_in,
                              void* d_out, int out_size, void* d_ws, size_t ws_size,
                              hipStream_t stream) {
  const float* logits  = (const float*)d_in[0];   // [16,19,512,1024] f32
  const int*   target  = (const int*)d_in[1];     // [16,512,1024] int (harness int32)
  unsigned int* ghist  = (unsigned int*)d_ws;     // 361 u32 bins
  float*       out     = (float*)d_out;           // scalar loss

  zero_hist<<<1, 512, 0, stream>>>(ghist);
  argmax_hist<<<kN4 / 256, 256, 0, stream>>>(logits, target, ghist);
  cm_loss<<<1, 32, 0, stream>>>(ghist, out);
}